// MLP_18528488915324
// MI455X (gfx1250) — compile-verified
//
#include <hip/hip_runtime.h>

// -------- CDNA5 wave32 vector types --------
typedef float v2f __attribute__((ext_vector_type(2)));
typedef float v4f __attribute__((ext_vector_type(4)));
typedef float v8f __attribute__((ext_vector_type(8)));

#define D_ELEMS (1u << 20)        // 1048576
#define N_ROWS_OUT 50
#define ROW_TILES 4               // 4 tiles x 16 rows covers 50 rows (clamped)
#define CHUNK_K 512               // K elements per wave
#define CHUNKS (D_ELEMS / CHUNK_K)            // 2048
#define WAVES_PER_BLOCK 8                      // 256 threads / wave32
#define TOTAL_WAVES (ROW_TILES * CHUNKS)       // 8192
#define NBLOCKS (TOTAL_WAVES / WAVES_PER_BLOCK) // 1024

__global__ void zero_out_kernel(float* __restrict__ out) {
    int t = threadIdx.x;
    if (t < N_ROWS_OUT) out[t] = 0.0f;
}

// out[m] = sum_k fc[m][k] * x[k], via V_WMMA_F32_16X16X4_F32.
// A (16x4 f32): lanes 0-15 hold row M=lane, K-slots {0,1}; lanes 16-31 hold
// row M=lane-16, K-slots {2,3}. B (4x16 f32): row K broadcast across columns.
// One b128 load per lane feeds two WMMAs (8 K values per iteration).
__global__ __launch_bounds__(256) void
matvec_wmma_f32(const float* __restrict__ fc,
                const float* __restrict__ x,
                float* __restrict__ out) {
    const int lane  = threadIdx.x & 31;
    const int wave  = blockIdx.x * WAVES_PER_BLOCK + (threadIdx.x >> 5);
    const int tile  = wave / CHUNKS;   // 0..3  (16 rows each)
    const int chunk = wave % CHUNKS;   // 0..2047 (512 K each)
    const int hi    = lane >> 4;       // 0: lanes 0-15, 1: lanes 16-31

    int row = tile * 16 + (lane & 15);
    if (row > N_ROWS_OUT - 1) row = N_ROWS_OUT - 1;  // clamp; dup addrs coalesce

    const size_t k0 = (size_t)chunk * CHUNK_K;
    const float* ap = fc + (size_t)row * D_ELEMS + k0 + (hi ? 4 : 0);
    const float* bp = x + k0 + (hi ? 4 : 0);

    v8f c = {0.f, 0.f, 0.f, 0.f, 0.f, 0.f, 0.f, 0.f};

#pragma unroll 4
    for (int k = 0; k < CHUNK_K; k += 8) {
        // fc is streamed exactly once: non-temporal. x is reused: temporal.
        v4f a = __builtin_nontemporal_load((const v4f*)ap);
        v4f b = *(const v4f*)bp;

        // WMMA 1: K-slots = {k, k+1} (low half) / {k+4, k+5} (high half)
        v2f a01 = {a.x, a.y};
        v2f b01 = {b.x, b.y};
        c = __builtin_amdgcn_wmma_f32_16x16x4_f32(
                /*neg_a=*/false, a01, /*neg_b=*/false, b01,
                /*c_mod=*/(short)0, c, /*reuse_a=*/false, /*reuse_b=*/false);

        // WMMA 2: K-slots = {k+2, k+3} / {k+6, k+7}
        v2f a23 = {a.z, a.w};
        v2f b23 = {b.z, b.w};
        c = __builtin_amdgcn_wmma_f32_16x16x4_f32(
                false, a23, false, b23, (short)0, c, false, false);

        ap += 8;
        bp += 8;
    }

    // D layout (16x16 f32 C/D): VGPR r -> lanes 0-15: M=r, lanes 16-31: M=r+8.
    // All columns identical (B columns were replicated), so column 0 suffices:
    // lane 0 holds rows tile*16+0..7, lane 16 holds rows tile*16+8..15.
    if ((lane & 15) == 0) {
        const int base = tile * 16 + (hi ? 8 : 0);
#pragma unroll
        for (int r = 0; r < 8; ++r) {
            int orow = base + r;
            if (orow < N_ROWS_OUT) {
                __hip_atomic_fetch_add(&out[orow], c[r],
                                       __ATOMIC_RELAXED,
                                       __HIP_MEMORY_SCOPE_AGENT);
            }
        }
    }
}

extern "C" void kernel_launch(void* const* d_in, const int* in_sizes, int n_in,
                              void* d_out, int out_size, void* d_ws, size_t ws_size,
                              hipStream_t stream) {
    (void)in_sizes; (void)n_in; (void)d_ws; (void)ws_size; (void)out_size;
    const float* x  = (const float*)d_in[0];  // "input": [D] f32
    const float* fc = (const float*)d_in[1];  // "fc": [50, D] f32
    float* out = (float*)d_out;               // [50] f32

    zero_out_kernel<<<1, 64, 0, stream>>>(out);
    matvec_wmma_f32<<<NBLOCKS, 256, 0, stream>>>(fc, x, out);
}